// snn_linear_26895085207935
// MI455X (gfx1250) — compile-verified
//
#include <hip/hip_runtime.h>

// ---------------------------------------------------------------------------
// SNN LIF layer on MI455X (gfx1250).
//
//   Phase 0 (cvt):  spikes int32 -> f32 staged in workspace (done once, 2 MB)
//   Phase 1 (GEMM, WMMA f32 16x16x4):  Y[t][n] = sum_k S[t][k] * W[n][k]
//   Phase 2 (scan): per-neuron LIF recurrence over t, emit spikes + mempot.
//
// d_in[0]: spikes_in, [64, 8192] boolean events, assumed delivered as int32 0/1
// d_in[1]: weights,   [8192(out), 8192(in)] float32, row-major
// d_out  : [65*8192] spike raster as float (row 0 zero) ++ [8192] final mempot
// d_ws   : [64*8192] f32 staged spikes ++ ksplit * [64*8192] f32 Y partials
// ---------------------------------------------------------------------------

typedef __attribute__((ext_vector_type(2))) float v2f;
typedef __attribute__((ext_vector_type(4))) float v4f;
typedef __attribute__((ext_vector_type(8))) float v8f;
typedef __attribute__((ext_vector_type(4))) int   v4i;

#define TSTEPS_  64
#define NEU_IN_  8192
#define NEU_OUT_ 8192
#define THRES_   1.0f
#define DECAY_   0.9375f

// ---------------------------------------------------------------------------
// Phase 0: one-shot int32 -> f32 conversion of the spike raster (b128 in/out).
// ---------------------------------------------------------------------------
__global__ __launch_bounds__(256)
void snn_cvt(const int* __restrict__ si, float* __restrict__ sf)
{
  const long i = 4l * (blockIdx.x * blockDim.x + threadIdx.x);
  const v4i v = *(const v4i*)(si + i);
  v4f f;
  f.x = (float)v.x; f.y = (float)v.y; f.z = (float)v.z; f.w = (float)v.w;
  *(v4f*)(sf + i) = f;
}

// ---------------------------------------------------------------------------
// Phase 1 GEMM: grid.x = NEU_OUT/64 (64-neuron strip per block, 16 per wave),
//       grid.y = K-split chunk. block = 128 threads = 4 waves (wave32).
// A (16x4, M=t):  lane L<16 holds M=L, K={0,1}; lane L>=16 holds M=L-16, K={2,3}
// B (4x16, N=n):  lane L<16 holds N=L, K={0,1}; lane L>=16 holds N=L-16, K={2,3}
// C/D (16x16 f32): VGPR v, lanes 0-15 -> M=v, lanes 16-31 -> M=v+8; N = lane&15
// A_IS_F32: A operand pre-staged as f32 (fast path, no VALU in the chain),
//           else raw int spikes with inline v_cvt (workspace-starved fallback).
// ---------------------------------------------------------------------------
template<bool A_IS_F32>
__global__ __launch_bounds__(128)
void snn_gemm_f32_wmma(const void* __restrict__ Araw,    // [64, 8192] f32 or i32
                       const float* __restrict__ W,      // [8192, 8192]
                       float* __restrict__ Y,            // [ksplit][64][8192]
                       int kchunk)                       // NEU_IN / ksplit
{
  const int lane  = threadIdx.x & 31;
  const int wave  = threadIdx.x >> 5;
  const int n0    = blockIdx.x * 64 + wave * 16;   // neuron strip for this wave
  const int lrow  = lane & 15;
  const int khalf = (lane >> 4) << 1;              // 0 (low half) or 2 (high half)
  const int kbase = blockIdx.y * kchunk;

  // per-lane streaming pointers (8-byte aligned: khalf even, k multiple of 4)
  const float* wp  = W + (long)(n0 + lrow) * NEU_IN_ + kbase + khalf;
  const long   abase = (long)lrow * NEU_IN_ + kbase + khalf;
  const float* af = (const float*)Araw + abase;
  const int*   ai = (const int*)Araw + abase;

  v8f acc[4] = {};   // 4 M-tiles (t = mt*16 .. mt*16+15) x 16 neurons

  for (int k = 0; k < kchunk; k += 32) {
    // pull the weight stream ahead (~4 KB ahead per row)
    __builtin_prefetch(wp + k + 1024, 0, 3);
#pragma unroll
    for (int kk = 0; kk < 32; kk += 4) {
      const v2f b = *(const v2f*)(wp + k + kk);    // W[n0+lrow][k_pair]
#pragma unroll
      for (int mt = 0; mt < 4; ++mt) {
        const long aoff = (long)mt * 16 * NEU_IN_ + k + kk;
        v2f a;
        if (A_IS_F32) {
          a = *(const v2f*)(af + aoff);            // pure b64 load, no VALU
        } else {
          const int* sq = ai + aoff;
          a.x = (float)sq[0];
          a.y = (float)sq[1];
        }
        // D = A(16x4) * B(4x16) + C, full-precision f32 matrix op
        acc[mt] = __builtin_amdgcn_wmma_f32_16x16x4_f32(
            /*neg_a=*/false, a, /*neg_b=*/false, b,
            /*c_mod=*/(short)0, acc[mt],
            /*reuse_a=*/false, /*reuse_b=*/false);
      }
    }
  }

  // Scatter accumulators: t = mt*16 + v + 8*(lane>=16), n = n0 + (lane&15)
  float* Yp = Y + (long)blockIdx.y * TSTEPS_ * NEU_OUT_;
  const int thi = (lane >> 4) * 8;
#pragma unroll
  for (int mt = 0; mt < 4; ++mt) {
#pragma unroll
    for (int v = 0; v < 8; ++v) {
      const int t = mt * 16 + thi + v;
      Yp[(long)t * NEU_OUT_ + n0 + lrow] = acc[mt][v];
    }
  }
}

// ---------------------------------------------------------------------------
// Phase 2 LIF scan: one thread per output neuron, coalesced column reads of Y.
//   m += Y[t][n]; spike = m >= 1; m = spike ? m-1 : m*DECAY
// ---------------------------------------------------------------------------
__global__ __launch_bounds__(256)
void snn_scan(const float* __restrict__ Y, float* __restrict__ out, int ksplit)
{
  const int n = blockIdx.x * blockDim.x + threadIdx.x;
  if (n >= NEU_OUT_) return;

  out[n] = 0.0f;                                   // spikes row 0 stays zero
  float m = 0.0f;
  for (int t = 0; t < TSTEPS_; ++t) {
    float inc = 0.0f;
    for (int c = 0; c < ksplit; ++c)
      inc += Y[((long)c * TSTEPS_ + t) * NEU_OUT_ + n];
    m += inc;
    const bool s = (m >= THRES_);
    out[(long)(t + 1) * NEU_OUT_ + n] = s ? 1.0f : 0.0f;
    m = s ? (m - THRES_) : (m * DECAY_);
  }
  out[(long)(TSTEPS_ + 1) * NEU_OUT_ + n] = m;     // final membrane potential
}

// ---------------------------------------------------------------------------
extern "C" void kernel_launch(void* const* d_in, const int* in_sizes, int n_in,
                              void* d_out, int out_size, void* d_ws, size_t ws_size,
                              hipStream_t stream)
{
  const int*   spikes = (const int*)d_in[0];
  const float* W      = (const float*)d_in[1];
  float*       out    = (float*)d_out;

  const long   selems = (long)TSTEPS_ * NEU_IN_;                       // 524288
  const size_t sbytes = (size_t)selems * sizeof(float);                // 2 MB
  const size_t ybytes = (size_t)TSTEPS_ * NEU_OUT_ * sizeof(float);    // 2 MB

  if (ws_size >= sbytes + ybytes) {
    // Fast path: stage spikes as f32 once, then pure-f32 WMMA GEMM.
    float* Sf = (float*)d_ws;
    float* Y  = Sf + selems;
    const size_t yroom = ws_size - sbytes;
    int ksplit = 1;
    if (yroom >= 4 * ybytes)      ksplit = 4;   // 512 blocks -> WGP coverage
    else if (yroom >= 2 * ybytes) ksplit = 2;
    const int kchunk = NEU_IN_ / ksplit;

    snn_cvt<<<selems / (256 * 4), 256, 0, stream>>>(spikes, Sf);
    dim3 grid(NEU_OUT_ / 64, ksplit);
    snn_gemm_f32_wmma<true><<<grid, 128, 0, stream>>>(Sf, W, Y, kchunk);
    snn_scan<<<NEU_OUT_ / 256, 256, 0, stream>>>(Y, out, ksplit);
  } else {
    // Fallback: inline int->f32 conversion inside the GEMM.
    float* Y = (float*)d_ws;
    dim3 grid(NEU_OUT_ / 64, 1);
    snn_gemm_f32_wmma<false><<<grid, 128, 0, stream>>>(spikes, W, Y, NEU_IN_);
    snn_scan<<<NEU_OUT_ / 256, 256, 0, stream>>>(Y, out, 1);
  }
}